// Path_6846177869886
// MI455X (gfx1250) — compile-verified
//
#include <hip/hip_runtime.h>

// MI455X / gfx1250 implementation.
// Dominant math: per-p, per-layer GEMM out[c,y] = sum_{h,x} (w1[h,c]*g[c,x]) * pair[h,y,x]
// via V_WMMA_F32_16X16X4_F32 (f32 in/out, matches reference precision), with
// double-buffered GLOBAL_LOAD_ASYNC_TO_LDS_B128 staging of pair[p,h] tiles.

typedef float v2f __attribute__((ext_vector_type(2)));
typedef float v8f __attribute__((ext_vector_type(8)));

#define WMMA_F32X4(a, b, c) \
    __builtin_amdgcn_wmma_f32_16x16x4_f32(false, (a), false, (b), (short)0, (c), false, false)

#define PP 128     // particles
#define NN 128     // constituents
#define TT 32      // pair channels
#define CC 128     // features
#define PITCH 132  // LDS row pitch in dwords: 132*4=528B rows -> 16B aligned, bank stride 4

// ---- async global->LDS staging (gfx1250 ASYNCcnt path), with sync fallback ----
#if defined(__has_builtin)
#if __has_builtin(__builtin_amdgcn_global_load_async_to_lds_b128)
#define HAVE_ASYNC_LDS 1
#endif
#endif

#if defined(HAVE_ASYNC_LDS)
// builtin parameter types per hipcc diagnostic: int4 (vector_size 16) pointers,
// global (AS1) source and LDS (AS3) destination, non-const.
typedef int b128_t __attribute__((vector_size(16)));
typedef __attribute__((address_space(1))) b128_t* g_b128_p;
typedef __attribute__((address_space(3))) b128_t* l_b128_p;

#if __has_builtin(__builtin_amdgcn_s_wait_asynccnt)
#define WAIT_ASYNC0() __builtin_amdgcn_s_wait_asynccnt(0)
#else
#define WAIT_ASYNC0() asm volatile("s_wait_asynccnt 0x0" ::: "memory")
#endif

// stage 64KB pair tile (128x128 f32, row-major) into pitched LDS buffer
__device__ __forceinline__ void stage_pair_async(const float* __restrict__ gsrc,
                                                 float* __restrict__ ldst, int tid) {
#pragma unroll
    for (int it = 0; it < 16; ++it) {
        const int e = it * 1024 + tid * 4;  // 256 thr * 4 floats = 1024/iter
        const int y = e >> 7, x = e & (NN - 1);
        __builtin_amdgcn_global_load_async_to_lds_b128(
            (g_b128_p)(gsrc + e), (l_b128_p)(ldst + y * PITCH + x), 0, 0);
    }
}
#endif

// synchronous staging fallback (also used by non-async builds)
__device__ __forceinline__ void stage_pair_sync(const float* __restrict__ gsrc,
                                                float* __restrict__ ldst, int tid) {
#pragma unroll
    for (int it = 0; it < 16; ++it) {
        const int e = it * 1024 + tid * 4;
        const int y = e >> 7, x = e & (NN - 1);
        float4 v = *(const float4*)(gsrc + e);
        *(float4*)(&ldst[y * PITCH + x]) = v;
    }
}

// ---------------------------------------------------------------------------
// Phase 1: wm, feat = pair @ wm, Qi = column-sum(pair), features = W^T @ feat
// ---------------------------------------------------------------------------
__global__ __launch_bounds__(256) void k_phase1(const float* __restrict__ pw,
                                                const float* __restrict__ pm,
                                                const float* __restrict__ pair,
                                                const float* __restrict__ weight,
                                                float* __restrict__ featBuf,
                                                float* __restrict__ qiBuf) {
    __shared__ float wm_s[NN];
    __shared__ float feat_s[TT * NN];  // 16 KB
    const int p = blockIdx.x;
    const int tid = threadIdx.x;

    if (tid < NN) wm_s[tid] = pw[p * NN + tid] * pm[p * NN + tid];
    __syncthreads();

    const float* pp = pair + (size_t)p * TT * NN * NN;

    // feat[t,i] = dot(pair[t,i,:], wm)
    for (int r = tid; r < TT * NN; r += 256) {
        const float* row = pp + (size_t)r * NN;
        float acc = 0.f;
#pragma unroll 4
        for (int j = 0; j < NN; j += 4) {
            float4 v = *(const float4*)(row + j);
            acc += v.x * wm_s[j] + v.y * wm_s[j + 1] + v.z * wm_s[j + 2] + v.w * wm_s[j + 3];
        }
        feat_s[r] = acc;
    }

    // Qi[t,x] = sum_i pair[t,i,x]  (lanes cover contiguous x -> coalesced)
    {
        const int x = tid & (NN - 1);
        for (int t = (tid >> 7); t < TT; t += 2) {
            const float* col = pp + (size_t)t * NN * NN + x;
            float s = 0.f;
#pragma unroll 8
            for (int i = 0; i < NN; ++i) s += col[i * NN];
            qiBuf[((size_t)p * TT + t) * NN + x] = s;
        }
    }
    __syncthreads();

    // features[c,i] = sum_t weight[t,c] * feat[t,i]   (M=C, N=NN, K=TT via WMMA f32)
    const int lane = tid & 31, wv = tid >> 5;
    const int l16 = lane & 15, half = lane >> 4, half2 = half << 1;
    const int c0 = wv * 16;

    v2f afrag[8];
#pragma unroll
    for (int ks = 0; ks < 8; ++ks) {
        const int k = ks * 4 + half2;
        afrag[ks].x = weight[k * CC + c0 + l16];
        afrag[ks].y = weight[(k + 1) * CC + c0 + l16];
    }

    float* fout = featBuf + (size_t)p * CC * NN;
#pragma unroll
    for (int it = 0; it < 8; ++it) {
        const int i0 = it * 16;
        v8f acc = {};
#pragma unroll
        for (int ks = 0; ks < 8; ++ks) {
            const int k = ks * 4 + half2;
            v2f b;
            b.x = feat_s[k * NN + i0 + l16];
            b.y = feat_s[(k + 1) * NN + i0 + l16];
            acc = WMMA_F32X4(afrag[ks], b, acc);
        }
        const int ch = half * 8;
#pragma unroll
        for (int r = 0; r < 8; ++r)
            fout[(c0 + ch + r) * NN + i0 + l16] = acc[r];
    }
}

// ---------------------------------------------------------------------------
// fc[p,c] = sum_x features[p,c,x] * wm[p,x]
// ---------------------------------------------------------------------------
__global__ __launch_bounds__(256) void k_fc(const float* __restrict__ pw,
                                            const float* __restrict__ pm,
                                            const float* __restrict__ featBuf,
                                            float* __restrict__ fcBuf) {
    __shared__ float red[256];
    const int p = blockIdx.x, tid = threadIdx.x;
    const int c = tid & (CC - 1), part = tid >> 7;
    const float* f = featBuf + (size_t)p * CC * NN + (size_t)c * NN;
    const float* w0 = pw + p * NN;
    const float* m0 = pm + p * NN;
    float s = 0.f;
    for (int x = part * 64; x < part * 64 + 64; ++x) s += f[x] * w0[x] * m0[x];
    red[tid] = s;
    __syncthreads();
    if (part == 0) fcBuf[p * CC + c] = red[c] + red[c + CC];
}

// ---------------------------------------------------------------------------
// stats: mean / rstd over P (ddof=1), stats[0:128]=mean, stats[128:256]=rstd
// ---------------------------------------------------------------------------
__global__ __launch_bounds__(128) void k_stats(const float* __restrict__ fcBuf,
                                               float* __restrict__ stats) {
    const int c = threadIdx.x;
    float s = 0.f, s2 = 0.f;
    for (int p = 0; p < PP; ++p) {
        const float v = fcBuf[p * CC + c];
        s += v;
        s2 += v * v;
    }
    const float mean = s * (1.f / (float)PP);
    const float var = (s2 - (float)PP * mean * mean) * (1.f / (float)(PP - 1));
    stats[c] = mean;
    stats[CC + c] = rsqrtf(var + 1e-5f);
}

// ---------------------------------------------------------------------------
// One HigherPoint layer (per-p workgroup):
//   f = (features - mean)*rstd*wm ; A = w0^T @ Qi ; g = f*A (LDS)
//   out[c,y] = sum_h sum_x (w1[h,c]*g[c,x]) * pair[p,h,y,x]   <- WMMA f32, K=T*N
//   features = relu(out + bias) + features(old)
// pair tiles double-buffered via async global->LDS DMA when available.
// ---------------------------------------------------------------------------
__global__ __launch_bounds__(256) void k_layer(const float* __restrict__ pw,
                                               const float* __restrict__ pm,
                                               const float* __restrict__ pair,
                                               const float* __restrict__ qiBuf,
                                               const float* __restrict__ hp_w0,
                                               const float* __restrict__ hp_w1,
                                               const float* __restrict__ hp_bias,
                                               const float* __restrict__ stats,
                                               float* __restrict__ featBuf,
                                               int l) {
    __shared__ float g_s[CC * PITCH];  // 67584 B, g[c,x] pitched
    __shared__ float qi_s[TT * NN];    // 16 KB
#if defined(HAVE_ASYNC_LDS)
    __shared__ float pbuf_s[2 * NN * PITCH];  // 135168 B, ping/pong halves by offset
#else
    __shared__ float pbuf_s[NN * PITCH];  // 67584 B, f_norm then pair staging
#endif

    const int p = blockIdx.x, tid = threadIdx.x;
    const int lane = tid & 31, wv = tid >> 5;
    const int l16 = lane & 15, half = lane >> 4, half2 = half << 1;
    const int c0 = wv * 16;
    const size_t PN = (size_t)CC * NN;

    float* fp = featBuf + (size_t)p * PN;

    // normalized, weighted features -> pbuf_s (first half) ; Qi -> qi_s
    for (int e = tid; e < CC * NN; e += 256) {
        const int c = e >> 7, x = e & (NN - 1);
        const float wmx = pw[p * NN + x] * pm[p * NN + x];
        pbuf_s[c * PITCH + x] = (fp[e] - stats[c]) * stats[CC + c] * wmx;
    }
    for (int e = tid; e < TT * NN; e += 256)
        qi_s[e] = qiBuf[(size_t)p * TT * NN + e];
    __syncthreads();

    // A[c,x] = sum_t hp_w0[l,t,c] * Qi[t,x]  (WMMA, K=32), then g = f_norm * A
    const float* w0l = hp_w0 + (size_t)l * TT * CC;
    v2f afrag[8];
#pragma unroll
    for (int ks = 0; ks < 8; ++ks) {
        const int k = ks * 4 + half2;
        afrag[ks].x = w0l[k * CC + c0 + l16];
        afrag[ks].y = w0l[(k + 1) * CC + c0 + l16];
    }
#pragma unroll
    for (int xt = 0; xt < 8; ++xt) {
        const int x0 = xt * 16;
        v8f acc = {};
#pragma unroll
        for (int ks = 0; ks < 8; ++ks) {
            const int k = ks * 4 + half2;
            v2f b;
            b.x = qi_s[k * NN + x0 + l16];
            b.y = qi_s[(k + 1) * NN + x0 + l16];
            acc = WMMA_F32X4(afrag[ks], b, acc);
        }
        const int ch = half * 8;
#pragma unroll
        for (int r = 0; r < 8; ++r) {
            const int c = c0 + ch + r, x = x0 + l16;
            g_s[c * PITCH + x] = pbuf_s[c * PITCH + x] * acc[r];
        }
    }
    __syncthreads();

    // main contraction: out[c,y] += (w1[h,c]*g[c,:]) @ pair[p,h,:,:]^T
    v8f out[8];
#pragma unroll
    for (int i = 0; i < 8; ++i) out[i] = (v8f){};

    const float* ppair = pair + (size_t)p * TT * PN;
    const float* w1l = hp_w1 + (size_t)l * TT * CC;

#if defined(HAVE_ASYNC_LDS)
    // double-buffered async pipeline: wait(h) -> barrier -> kick(h+1) -> compute(h)
    stage_pair_async(ppair, pbuf_s, tid);  // prologue: h = 0 -> half 0
    for (int h = 0; h < TT; ++h) {
        const float* cur = pbuf_s + (size_t)(h & 1) * (NN * PITCH);
        WAIT_ASYNC0();    // this wave's stage(h) DMA complete
        __syncthreads();  // all waves: stage(h) visible, compute(h-1) reads done
        if (h + 1 < TT)
            stage_pair_async(ppair + (size_t)(h + 1) * PN,
                             pbuf_s + (size_t)((h + 1) & 1) * (NN * PITCH), tid);

        const float sA = w1l[h * CC + c0 + l16];  // per-A-row scale (scalar per lane)
        for (int xk = 0; xk < NN; xk += 4) {
            const int k = xk + half2;
            const v2f ga = *(const v2f*)(&g_s[(c0 + l16) * PITCH + k]);
            v2f a;
            a.x = ga.x * sA;
            a.y = ga.y * sA;
#pragma unroll
            for (int yt = 0; yt < 8; ++yt) {
                const v2f b = *(const v2f*)(&cur[(yt * 16 + l16) * PITCH + k]);
                out[yt] = WMMA_F32X4(a, b, out[yt]);
            }
        }
    }
#else
    // synchronous single-buffer fallback
    for (int h = 0; h < TT; ++h) {
        __syncthreads();
        const float* ph = ppair + (size_t)h * PN;
        if (h + 1 < TT) __builtin_prefetch(ppair + (size_t)(h + 1) * PN, 0, 1);
        stage_pair_sync(ph, pbuf_s, tid);
        __syncthreads();

        const float sA = w1l[h * CC + c0 + l16];
        for (int xk = 0; xk < NN; xk += 4) {
            const int k = xk + half2;
            const v2f ga = *(const v2f*)(&g_s[(c0 + l16) * PITCH + k]);
            v2f a;
            a.x = ga.x * sA;
            a.y = ga.y * sA;
#pragma unroll
            for (int yt = 0; yt < 8; ++yt) {
                const v2f b = *(const v2f*)(&pbuf_s[(yt * 16 + l16) * PITCH + k]);
                out[yt] = WMMA_F32X4(a, b, out[yt]);
            }
        }
    }
#endif
    __syncthreads();

    // epilogue: features = relu(out + bias) + shortcut
    const float* bl = hp_bias + (size_t)l * CC;
    const int ch = half * 8;
#pragma unroll
    for (int yt = 0; yt < 8; ++yt) {
        const int y = yt * 16 + l16;
#pragma unroll
        for (int r = 0; r < 8; ++r) {
            const int c = c0 + ch + r;
            float v = out[yt][r] + bl[c];
            v = v > 0.f ? v : 0.f;
            fp[c * NN + y] = v + fp[c * NN + y];
        }
    }
}

// ---------------------------------------------------------------------------
// jet[p,c] = sum_x features[p,c,x] * wm[p,x]
// ---------------------------------------------------------------------------
__global__ __launch_bounds__(256) void k_jet(const float* __restrict__ pw,
                                             const float* __restrict__ pm,
                                             const float* __restrict__ featBuf,
                                             float* __restrict__ outp) {
    __shared__ float red[256];
    const int p = blockIdx.x, tid = threadIdx.x;
    const int c = tid & (CC - 1), part = tid >> 7;
    const float* f = featBuf + (size_t)p * CC * NN + (size_t)c * NN;
    const float* w0 = pw + p * NN;
    const float* m0 = pm + p * NN;
    float s = 0.f;
    for (int x = part * 64; x < part * 64 + 64; ++x) s += f[x] * w0[x] * m0[x];
    red[tid] = s;
    __syncthreads();
    if (part == 0) outp[p * CC + c] = red[c] + red[c + CC];
}

// ---------------------------------------------------------------------------
extern "C" void kernel_launch(void* const* d_in, const int* in_sizes, int n_in,
                              void* d_out, int out_size, void* d_ws, size_t ws_size,
                              hipStream_t stream) {
    const float* pw = (const float*)d_in[0];       // part_weight [P,N]
    const float* pm = (const float*)d_in[1];       // part_mask   [P,N]
    const float* pair = (const float*)d_in[2];     // pair_weight [P,T,N,N]
    const float* weight = (const float*)d_in[3];   // weight      [T,C]
    const float* hp_w0 = (const float*)d_in[4];    // [L,T,C]
    const float* hp_w1 = (const float*)d_in[5];    // [L,T,C]
    const float* hp_bias = (const float*)d_in[6];  // [L,C,1]
    float* out = (float*)d_out;                    // [P,C]

    float* ws = (float*)d_ws;
    float* featBuf = ws;                            // P*C*N   = 2,097,152 f32
    float* qiBuf = featBuf + (size_t)PP * CC * NN;  // P*T*N   =   524,288 f32
    float* fcBuf = qiBuf + (size_t)PP * TT * NN;    // P*C     =    16,384 f32
    float* stats = fcBuf + (size_t)PP * CC;         // 2*C     =       256 f32
    (void)ws_size; (void)in_sizes; (void)n_in; (void)out_size;

    k_phase1<<<PP, 256, 0, stream>>>(pw, pm, pair, weight, featBuf, qiBuf);

    for (int l = 0; l < 2; ++l) {
        k_fc<<<PP, 256, 0, stream>>>(pw, pm, featBuf, fcBuf);
        k_stats<<<1, 128, 0, stream>>>(fcBuf, stats);
        k_layer<<<PP, 256, 0, stream>>>(pw, pm, pair, qiBuf, hp_w0, hp_w1, hp_bias,
                                        stats, featBuf, l);
    }

    k_jet<<<PP, 256, 0, stream>>>(pw, pm, featBuf, out);
}